// GlobalAttn_30872224924156
// MI455X (gfx1250) — compile-verified
//
#include <hip/hip_runtime.h>
#include <hip/hip_bf16.h>

// Shapes from the reference
#define H 1024
#define B 64
#define L 2048

typedef __attribute__((ext_vector_type(2))) float v2f;
typedef __attribute__((ext_vector_type(4))) float f4;
typedef __attribute__((ext_vector_type(8))) float v8f;

// ---------------------------------------------------------------------------
// Kernel 1: v = hidden @ W   ([B,H] x [H,H] -> [B,H]), full fp32 via
// V_WMMA_F32_16X16X4_F32. One wave per 16x16 output tile; EXEC all-ones.
// A[m][k] = hidden[m*H+k]; B[k][n] = W[k*H+n] (W is [g,h] row-major, g=K).
// ---------------------------------------------------------------------------
__global__ __launch_bounds__(32) void vproj_wmma(const float* __restrict__ hid,
                                                 const float* __restrict__ W,
                                                 float* __restrict__ v) {
    const int mTile = blockIdx.x & 3;    // 4 tiles of M (B=64)
    const int nTile = blockIdx.x >> 2;   // 64 tiles of N (H=1024)
    const int lane  = threadIdx.x;
    const int half  = lane >> 4;         // 0: lanes 0-15, 1: lanes 16-31
    const int l16   = lane & 15;
    const int arow  = mTile * 16 + l16;  // A-matrix row owned by this lane
    const int ncol  = nTile * 16 + l16;  // B/D column owned by this lane

    v8f acc = {};
#pragma unroll 4
    for (int k = 0; k < H; k += 4) {
        const int kk = k + half * 2;     // lanes 16-31 handle K+2,K+3
        v2f a, b;
        a.x = hid[arow * H + kk];
        a.y = hid[arow * H + kk + 1];
        b.x = W[(kk) * H + ncol];
        b.y = W[(kk + 1) * H + ncol];
        acc = __builtin_amdgcn_wmma_f32_16x16x4_f32(
            /*neg_a=*/false, a, /*neg_b=*/false, b,
            /*c_mod=*/(short)0, acc, /*reuse_a=*/false, /*reuse_b=*/false);
    }
    // D layout: VGPR r -> row r (lanes 0-15) / row r+8 (lanes 16-31)
#pragma unroll
    for (int r = 0; r < 8; ++r) {
        const int m = mTile * 16 + r + half * 8;
        v[m * H + ncol] = acc[r];
    }
}

// ---------------------------------------------------------------------------
// Kernel 2: scores[b,l] = <enc[l,b,:], v[b,:]>. One wave per (l,b) row:
// contiguous 4KB streaming read of enc (non-temporal; enc >> L2), v stays hot
// in L2. This is the 512MB / 23.3TB/s ~ 22us bandwidth-bound phase.
// ---------------------------------------------------------------------------
__global__ __launch_bounds__(256) void scores_dot(const float* __restrict__ enc,
                                                  const float* __restrict__ v,
                                                  float* __restrict__ scores) {
    const int wave = (int)((blockIdx.x * blockDim.x + threadIdx.x) >> 5);
    const int lane = threadIdx.x & 31;
    const int l = wave >> 6;   // L index
    const int b = wave & 63;   // B index

    const f4* __restrict__ erow = (const f4*)(enc + ((size_t)l * B + b) * H);
    const f4* __restrict__ vrow = (const f4*)(v + (size_t)b * H);

    float acc = 0.f;
#pragma unroll
    for (int i = 0; i < 8; ++i) {
        const int idx = lane + i * 32;           // 32 lanes x 8 x float4 = 1024
        f4 e = __builtin_nontemporal_load(&erow[idx]);
        f4 w = vrow[idx];
        acc += e.x * w.x + e.y * w.y + e.z * w.z + e.w * w.w;
    }
#pragma unroll
    for (int off = 16; off > 0; off >>= 1)
        acc += __shfl_down(acc, off, 32);
    if (lane == 0) scores[(size_t)b * L + l] = acc;
}

// ---------------------------------------------------------------------------
// Kernel 3: row softmax over L=2048 per batch. One 256-thread block per b.
// (The bias term <bias, h[b,:]> is a per-row constant -> softmax-invariant,
//  so it is intentionally omitted.)
// ---------------------------------------------------------------------------
__global__ __launch_bounds__(256) void softmax_rows(const float* __restrict__ scores,
                                                    float* __restrict__ out) {
    const int b = blockIdx.x;
    const int t = threadIdx.x;
    const int lane = t & 31;
    const int wid = t >> 5;
    __shared__ float sred[8];
    __shared__ float sval;

    const float* __restrict__ row = scores + (size_t)b * L;
    float vals[8];
    float m = -3.0e38f;
#pragma unroll
    for (int i = 0; i < 8; ++i) {
        vals[i] = row[t + i * 256];
        m = fmaxf(m, vals[i]);
    }
#pragma unroll
    for (int off = 16; off > 0; off >>= 1)
        m = fmaxf(m, __shfl_down(m, off, 32));
    if (lane == 0) sred[wid] = m;
    __syncthreads();
    if (t == 0) {
        float mm = sred[0];
#pragma unroll
        for (int i = 1; i < 8; ++i) mm = fmaxf(mm, sred[i]);
        sval = mm;
    }
    __syncthreads();
    m = sval;

    float e[8];
    float s = 0.f;
#pragma unroll
    for (int i = 0; i < 8; ++i) {
        e[i] = __expf(vals[i] - m);
        s += e[i];
    }
#pragma unroll
    for (int off = 16; off > 0; off >>= 1)
        s += __shfl_down(s, off, 32);
    if (lane == 0) sred[wid] = s;
    __syncthreads();
    if (t == 0) {
        float ss = 0.f;
#pragma unroll
        for (int i = 0; i < 8; ++i) ss += sred[i];
        sval = ss;
    }
    __syncthreads();
    const float inv = 1.0f / sval;
#pragma unroll
    for (int i = 0; i < 8; ++i)
        out[(size_t)b * L + t + i * 256] = e[i] * inv;
}

extern "C" void kernel_launch(void* const* d_in, const int* in_sizes, int n_in,
                              void* d_out, int out_size, void* d_ws, size_t ws_size,
                              hipStream_t stream) {
    const float* hid = (const float*)d_in[0];   // [B,H]
    const float* enc = (const float*)d_in[1];   // [L,B,H]
    const float* W   = (const float*)d_in[2];   // [H,H]
    // d_in[3] = bias [H]: softmax-invariant, unused.
    float* out = (float*)d_out;                 // [B,1,L] flat

    float* v      = (float*)d_ws;               // B*H floats   (256 KB)
    float* scores = v + (size_t)B * H;          // B*L floats   (512 KB)

    // 1) v = hidden @ W  (4 M-tiles x 64 N-tiles = 256 waves)
    vproj_wmma<<<4 * 64, 32, 0, stream>>>(hid, W, v);

    // 2) scores: one wave per (l,b) row -> L*B waves, 8 waves/block
    scores_dot<<<(L * B) / 8, 256, 0, stream>>>(enc, v, scores);

    // 3) softmax per batch row
    softmax_rows<<<B, 256, 0, stream>>>(scores, out);
}